// CropAndPadMask_36807869727194
// MI455X (gfx1250) — compile-verified
//
#include <hip/hip_runtime.h>
#include <hip/hip_bf16.h>

// Problem constants from the reference setup (B=4, N=64 derived from sizes at launch).
#define HH 512
#define WW 512
#define MHH 28
#define MWW 28

typedef float v2f __attribute__((ext_vector_type(2)));
typedef float v4f __attribute__((ext_vector_type(4)));
typedef float v8f __attribute__((ext_vector_type(8)));

static constexpr int MP = 33;   // padded LDS row stride (bank-conflict avoidance)
static constexpr int NW = 8;    // waves per 256-thread block (wave32)

__device__ __forceinline__ v8f wmma4(v2f a, v2f b, v8f c) {
  // V_WMMA_F32_16X16X4_F32 : D(16x16,f32) = A(16x4,f32) * B(4x16,f32) + C
  return __builtin_amdgcn_wmma_f32_16x16x4_f32(false, a, false, b, (short)0, c,
                                               false, false);
}

struct Interp { int i0, i1; float f; };

__device__ __forceinline__ Interp interp_coord(int g, int gmin, float s, int msz) {
  float src = (float)(g - gmin) * s;
  src = fminf(fmaxf(src, 0.0f), (float)(msz - 1));
  Interp r;
  r.i0 = (int)floorf(src);
  r.i1 = min(r.i0 + 1, msz - 1);
  r.f  = src - (float)r.i0;
  return r;
}

// Interpolation-matrix entry: weight of mask index j for this output coordinate.
__device__ __forceinline__ float wgt(const Interp& it, int j) {
  float w = 0.0f;
  if (j == it.i0) w += 1.0f - it.f;
  if (j == it.i1) w += it.f;
  return w;
}

__global__ __launch_bounds__(256)
void paste_masks_kernel(const float* __restrict__ det,
                        const float* __restrict__ mask,
                        float* __restrict__ out, int BN) {
  __shared__ float Mpad[32 * MP];        // 28x28 mask zero-padded to 32x32
  __shared__ float Rbuf[NW][16 * MP];    // per-wave GEMM1 -> GEMM2 staging
  __shared__ float red[256];             // score-max reduction

  const int blk = blockIdx.x;            // blk = b*N + n
  const int tid = threadIdx.x;
  const float* msk = mask + (size_t)blk * (MHH * MWW);
  __builtin_prefetch(msk, 0, 3);         // global_prefetch_b8

  // ---- stage mask into LDS (zero-padded to 32x32, stride 33) ----
  for (int i = tid; i < 32 * MP; i += blockDim.x) {
    int r = i / MP, c = i - r * MP;
    Mpad[i] = (r < MHH && c < MWW) ? msk[r * MWW + c] : 0.0f;
  }

  // ---- stream zeros over this (b,n) slice: the bandwidth floor (~11us total) ----
  float* oslice = out + (size_t)blk * (HH * WW);
  v4f zero4 = {0.f, 0.f, 0.f, 0.f};
  v4f* o4 = (v4f*)oslice;
  for (int i = tid; i < (HH * WW) / 4; i += blockDim.x)
    __builtin_nontemporal_store(zero4, &o4[i]);   // output is never re-read

  // ---- global threshold: thr = max(all scores) > 50 ? 50 : -100 ----
  float mx = -3.0e38f;
  for (int i = tid; i < BN; i += blockDim.x) mx = fmaxf(mx, det[i * 6 + 5]);
  red[tid] = mx;
  __syncthreads();
  for (int s = 128; s > 0; s >>= 1) {
    if (tid < s) red[tid] = fmaxf(red[tid], red[tid + s]);
    __syncthreads();
  }
  const float thr = (red[0] > 50.0f) ? 50.0f : -100.0f;

  // ---- box parameters (block-uniform -> scalar regs) ----
  const float* d = det + blk * 6;
  const float score = d[5];
  const float cx = fmaxf(d[0], 1.0f), cy = fmaxf(d[1], 1.0f);
  const float bw = fmaxf(d[2], 1.0f), bh = fmaxf(d[3], 1.0f);
  const int xmin = min(max((int)ceilf(cx - 0.5f * bw), 0), WW);
  const int xmax = min(max((int)ceilf(cx + 0.5f * bw), 0), WW);
  const int ymin = min(max((int)ceilf(cy - 0.5f * bh), 0), HH);
  const int ymax = min(max((int)ceilf(cy + 0.5f * bh), 0), HH);
  const int oh = ymax - ymin, ow = xmax - xmin;
  const float sy = (oh > 1) ? (float)(MHH - 1) / fmaxf((float)oh - 1.0f, 1.0f) : 0.0f;
  const float sx = (ow > 1) ? (float)(MWW - 1) / fmaxf((float)ow - 1.0f, 1.0f) : 0.0f;
  const bool valid = (score >= thr) && (oh > 0) && (ow > 0);

  // All waves' zero-stores must complete before any wave overwrites the box
  // region (cross-wave WAW): drain STOREcnt, then workgroup barrier.
  asm volatile("s_wait_storecnt 0x0" ::: "memory");
  __syncthreads();
  if (!valid) return;

  // ---- box region as 16x16 WMMA tiles: OUT = Wy * Mpad * Wx^T ----
  const int ntx = (ow + 15) >> 4, nty = (oh + 15) >> 4;
  const int ntiles = ntx * nty;
  const int wid = tid >> 5, lane = tid & 31;
  const int lo = lane & 15;
  const bool hi = lane >= 16;
  float* rb = &Rbuf[wid][0];

  for (int t = wid; t < ntiles; t += NW) {       // wave-uniform: EXEC all-ones
    const int ty = t / ntx, tx = t - ty * ntx;
    const int yg0 = ymin + ty * 16, xg0 = xmin + tx * 16;
    const Interp iy = interp_coord(yg0 + lo, ymin, sy, MHH);
    const Interp ix = interp_coord(xg0 + lo, xmin, sx, MWW);

    // GEMM1: R(16x32) = Wy(16x32) x Mpad(32x32), K-steps of 4
    v8f r0 = {}; v8f r1 = {};
    #pragma unroll
    for (int k0 = 0; k0 < 8; ++k0) {
      const int kb = 4 * k0 + (hi ? 2 : 0);
      v2f a; a.x = wgt(iy, kb); a.y = wgt(iy, kb + 1);       // A: on-the-fly weights
      v2f b0; b0.x = Mpad[kb * MP + lo];       b0.y = Mpad[(kb + 1) * MP + lo];
      v2f b1; b1.x = Mpad[kb * MP + lo + 16];  b1.y = Mpad[(kb + 1) * MP + lo + 16];
      r0 = wmma4(a, b0, r0);
      r1 = wmma4(a, b1, r1);
    }

    // Re-layout R (C/D layout) -> row-major LDS so it can feed GEMM2 as A.
    // DS ops are in-order within a wave: no explicit wait needed.
    {
      const int mb = hi ? 8 : 0;
      #pragma unroll
      for (int r = 0; r < 8; ++r) {
        rb[(mb + r) * MP + lo]      = r0[r];
        rb[(mb + r) * MP + lo + 16] = r1[r];
      }
    }

    // GEMM2: OUT(16x16) = R(16x32) x Wx^T(32x16)
    v8f acc = {};
    #pragma unroll
    for (int k0 = 0; k0 < 8; ++k0) {
      const int kb = 4 * k0 + (hi ? 2 : 0);
      v2f a; a.x = rb[lo * MP + kb]; a.y = rb[lo * MP + kb + 1];
      v2f b; b.x = wgt(ix, kb);      b.y = wgt(ix, kb + 1);
      acc = wmma4(a, b, acc);
    }

    // Scatter tile (C/D layout), predicated on box edges (post-WMMA: legal).
    const int mb = hi ? 8 : 0;
    const int xg = xg0 + lo;
    #pragma unroll
    for (int r = 0; r < 8; ++r) {
      const int yg = yg0 + mb + r;
      if (yg < ymax && xg < xmax)
        __builtin_nontemporal_store(acc[r], &oslice[yg * WW + xg]);
    }
  }
}

extern "C" void kernel_launch(void* const* d_in, const int* in_sizes, int n_in,
                              void* d_out, int out_size, void* d_ws, size_t ws_size,
                              hipStream_t stream) {
  // d_in[0]: images [B,H,W,3] (only shapes matter), d_in[1]: det_outs [B,N,6],
  // d_in[2]: ins_outs [B,N,28,28]. Output: [B,N,H,W] f32.
  const float* det = (const float*)d_in[1];
  const float* ins = (const float*)d_in[2];
  float* out = (float*)d_out;
  const int BN = in_sizes[1] / 6;   // B*N (= 256 for the reference setup)
  (void)d_ws; (void)ws_size; (void)n_in; (void)out_size;

  paste_masks_kernel<<<BN, 256, 0, stream>>>(det, ins, out, BN);
}